// TfidfWeightedMeanPooling_80169859547707
// MI455X (gfx1250) — compile-verified
//
#include <hip/hip_runtime.h>
#include <hip/hip_bf16.h>

// TF-IDF weighted mean pooling, MI455X (gfx1250, wave32).
// B=32, L=2048, D=768, V=32000.
// Bandwidth-bound: 201MB of f32 hidden states -> ~8.6us floor at 23.3 TB/s.
// Einsum done with V_WMMA_F32_16X16X4_F32 (exact f32 A/B/C, RNE).

#define Bv 32
#define Lv 2048
#define Dv 768
#define Vv 32000

typedef __attribute__((ext_vector_type(2))) float v2f;
typedef __attribute__((ext_vector_type(8))) float v8f;

// ---------------------------------------------------------------------------
// Kernel 1: per-batch histogram -> per-token weights and 1/norm.
// One 1024-thread workgroup per batch row. Histogram packed 2x u16 per u32
// (max count 2048 < 65536, so no cross-half carry) -> 64,000 B of LDS.
// ---------------------------------------------------------------------------
__global__ __launch_bounds__(1024) void tfidf_weights_kernel(
    const int* __restrict__ mask,      // [B,L]  int32
    const int* __restrict__ ids,       // [B,L]  int32 (JAX default x64-off)
    const float* __restrict__ idf,     // [V]
    const float* __restrict__ alpha_p, // [1]
    float* __restrict__ wts,           // [B,L]  out
    float* __restrict__ inv_norm)      // [B]    out
{
    __shared__ unsigned int hist[Vv / 2];   // packed: entry v>>1, half v&1
    __shared__ float fred[32];              // one partial per wave
    __shared__ unsigned int nvalid_s;

    const int b   = blockIdx.x;
    const int tid = threadIdx.x;
    const int lane = tid & 31;
    const int wave = tid >> 5;

    for (int i = tid; i < Vv / 2; i += 1024) hist[i] = 0u;
    if (tid == 0) nvalid_s = 0u;
    __syncthreads();

    // Pass 1: masked histogram + n_valid (integer atomics -> deterministic).
    unsigned int mycnt = 0;
    for (int l = tid; l < Lv; l += 1024) {
        int m = mask[b * Lv + l];
        if (m != 0) {
            int id = ids[b * Lv + l];
            atomicAdd(&hist[id >> 1], 1u << ((id & 1) * 16));
            mycnt += 1u;
        }
    }
    if (mycnt) atomicAdd(&nvalid_s, mycnt);
    __syncthreads();

    const float nv    = fmaxf((float)nvalid_s, 1.0f);
    const float alpha = *alpha_p;

    // Pass 2: weights = (1 + alpha*tf*idf)*mask ; accumulate weight sum.
    float wsum = 0.0f;
    for (int l = tid; l < Lv; l += 1024) {
        int id  = ids[b * Lv + l];
        float m = (float)mask[b * Lv + l];
        unsigned int cnt = (hist[id >> 1] >> ((id & 1) * 16)) & 0xFFFFu;
        float tf = (float)cnt / nv;
        float w  = (1.0f + alpha * tf * idf[id]) * m;
        wts[b * Lv + l] = w;
        wsum += w;
    }

    // Deterministic reduction: wave shuffle tree, then wave 0 sums 32 partials
    // in a fixed order.
    for (int off = 16; off > 0; off >>= 1)
        wsum += __shfl_down(wsum, off, 32);
    if (lane == 0) fred[wave] = wsum;
    __syncthreads();
    if (tid == 0) {
        float s = 0.0f;
        for (int wv = 0; wv < 32; ++wv) s += fred[wv];
        inv_norm[b] = 1.0f / fmaxf(s, 1e-12f);
    }
}

// ---------------------------------------------------------------------------
// Kernel 2: pooled[b, d0:d0+16] = (1/norm[b]) * sum_l w[b,l] * H[b,l,d0:d0+16]
// via V_WMMA_F32_16X16X4_F32. A row0 carries 4 weights (rows 1..15 zero),
// B = H[l0..l0+3, d0..d0+15]. C row0 (VGPR0, lanes 0-15) accumulates.
// Grid (48 d-tiles, 32 batches), 256 threads = 8 waves; each wave owns a
// 256-long L chunk -> 64 chained WMMAs, then fixed-order LDS reduction.
// ---------------------------------------------------------------------------
__global__ __launch_bounds__(256) void pool_wmma_kernel(
    const float* __restrict__ H,        // [B,L,D]
    const float* __restrict__ wts,      // [B,L]
    const float* __restrict__ inv_norm, // [B]
    float* __restrict__ out)            // [B,D]
{
    const int dt   = blockIdx.x;        // d-tile 0..47
    const int b    = blockIdx.y;        // batch 0..31
    const int tid  = threadIdx.x;
    const int wave = tid >> 5;          // 0..7
    const int lane = tid & 31;
    const int half = lane >> 4;         // 0: K slots {0,1}; 1: K slots {2,3}
    const int n    = lane & 15;         // N column / M row index
    const int d0   = dt * 16;

    const float* __restrict__ Hb = H + (size_t)b * Lv * Dv;
    const float* __restrict__ wb = wts + b * Lv;

    v8f c = {};                         // C/D accumulator, all-f32
    const int lbase = wave * (Lv / 8);  // 256-token chunk per wave

    #pragma unroll 4
    for (int kk = 0; kk < Lv / 8; kk += 4) {
        const int l0 = lbase + kk;
        // B tile: v0 = row K{0|2}, v1 = row K{1|3}; 16 contiguous floats/row.
        v2f bv;
        bv.x = Hb[(size_t)(l0 + 2 * half + 0) * Dv + d0 + n];
        bv.y = Hb[(size_t)(l0 + 2 * half + 1) * Dv + d0 + n];
        // A tile: M=0 row holds weights; all other rows zero.
        // Lane (n==0, half=0) -> K0,K1 ; lane (n==0, half=1) -> K2,K3.
        float w0 = wb[l0 + 2 * half + 0];   // uniform per half-wave, no branch
        float w1 = wb[l0 + 2 * half + 1];
        v2f av;
        av.x = (n == 0) ? w0 : 0.0f;
        av.y = (n == 0) ? w1 : 0.0f;
        // D = A*B + C  (8 args: neg_a, A, neg_b, B, c_mod, C, reuse_a, reuse_b)
        c = __builtin_amdgcn_wmma_f32_16x16x4_f32(
                false, av, false, bv, (short)0, c, false, false);
    }

    // C row 0 lives in c[0] on lanes 0..15 (N = lane). Reduce 8 waves.
    __shared__ float red[8][16];
    if (lane < 16) red[wave][n] = c[0];
    __syncthreads();
    if (tid < 16) {
        float s = 0.0f;
        #pragma unroll
        for (int wv = 0; wv < 8; ++wv) s += red[wv][tid];
        out[b * Dv + d0 + tid] = s * inv_norm[b];
    }
}

// ---------------------------------------------------------------------------
extern "C" void kernel_launch(void* const* d_in, const int* in_sizes, int n_in,
                              void* d_out, int out_size, void* d_ws, size_t ws_size,
                              hipStream_t stream) {
    const float* H      = (const float*)d_in[0];  // [B,L,D] f32
    const int*   mask   = (const int*)d_in[1];    // [B,L]   i32
    const int*   ids    = (const int*)d_in[2];    // [B,L]   i32
    const float* idf    = (const float*)d_in[3];  // [V]     f32
    const float* alpha  = (const float*)d_in[4];  // [1]     f32
    float*       out    = (float*)d_out;          // [B,D]   f32

    // Workspace: weights [B*L] floats, then inv_norm [B] floats (~256.1 KB).
    float* wts      = (float*)d_ws;
    float* inv_norm = wts + Bv * Lv;

    tfidf_weights_kernel<<<dim3(Bv), dim3(1024), 0, stream>>>(
        mask, ids, idf, alpha, wts, inv_norm);

    pool_wmma_kernel<<<dim3(Dv / 16, Bv), dim3(256), 0, stream>>>(
        H, wts, inv_norm, out);
}